// Aggregation_59966333386863
// MI455X (gfx1250) — compile-verified
//
#include <hip/hip_runtime.h>

#ifndef __has_builtin
#define __has_builtin(x) 0
#endif

#define AS1 __attribute__((address_space(1)))
#define AS3 __attribute__((address_space(3)))

// ---- CDNA5 async global->LDS copy (gfx1250) --------------------------------
__device__ __forceinline__ void async_ld_f32(const float* gp, float* lp) {
#if __has_builtin(__builtin_amdgcn_global_load_async_to_lds_b32)
  __builtin_amdgcn_global_load_async_to_lds_b32(
      (AS1 int*)gp, (AS3 int*)lp, /*offset=*/0, /*cpol=*/0);
#else
  *lp = *gp;  // synchronous fallback (still correct)
#endif
}

#if __has_builtin(__builtin_amdgcn_s_wait_asynccnt)
#define WAIT_ASYNC(n) __builtin_amdgcn_s_wait_asynccnt(n)
#elif __has_builtin(__builtin_amdgcn_global_load_async_to_lds_b32)
#define WAIT_ASYNC(n) asm volatile("s_wait_asynccnt %0" ::"i"(n) : "memory")
#else
#define WAIT_ASYNC(n) ((void)0)
#endif
// ----------------------------------------------------------------------------

namespace cfg {
constexpr int N = 8, C = 256, H = 128, W = 128;
constexpr int WC = 32, G = 8;            // 8 groups share each weight
constexpr int TH = 8, TW = 32;           // spatial tile per block (256 px)
constexpr int HR = TH + 2;               // 10 halo rows
constexpr int WR = TW + 2;               // 34 halo cols
constexpr int LSTRIDE = 36;              // padded LDS row stride (floats)
constexpr int HALO = HR * WR;            // 340 elements per tile
}  // namespace cfg

__device__ __forceinline__ int refl(int i, int n) {
  i = (i < 0) ? -i : i;                  // reflect (no edge repeat): -1 -> 1
  i = (i >= n) ? (2 * n - 2 - i) : i;    //                         : n -> n-2
  return i;
}

__global__ __launch_bounds__(256) void aggregation_kernel(
    const float* __restrict__ x, const float* __restrict__ wgt,
    float* __restrict__ out) {
  using namespace cfg;
  __shared__ float buf[2][HR * LSTRIDE];

  const int tx = threadIdx.x;            // 0..31
  const int ty = threadIdx.y;            // 0..7
  const int tid = ty * TW + tx;          // 0..255

  const int tile = blockIdx.x;           // 0..63  (16 h-tiles x 4 w-tiles)
  const int wc = blockIdx.y;             // 0..31
  const int n = blockIdx.z;              // 0..7

  const int w0 = (tile & 3) * TW;
  const int h0 = (tile >> 2) * TH;
  const int h = h0 + ty;
  const int w = w0 + tx;

  // Per-pixel weights: loaded once, reused for all 8 groups. Coalesced:
  // consecutive tx -> consecutive w -> contiguous 128B per wave per tap.
  float wt[9];
  {
    const float* wp =
        wgt + ((size_t)(n * WC + wc) * 9 * H + h) * W + w;
#pragma unroll
    for (int k = 0; k < 9; ++k) wt[k] = wp[(size_t)k * H * W];
  }

  // Each thread stages exactly 2 halo elements per tile (clamped so every
  // lane issues both async loads -> uniform per-wave ASYNCcnt accounting).
  int lofs[2], gofs[2];
#pragma unroll
  for (int j = 0; j < 2; ++j) {
    int i = tid + j * 256;
    i = (i < HALO) ? i : (HALO - 1);
    const int r = i / WR;
    const int cc = i - r * WR;
    lofs[j] = r * LSTRIDE + cc;
    gofs[j] = refl(h0 + r - 1, H) * W + refl(w0 + cc - 1, W);
  }

  const float* xn = x + (size_t)n * C * H * W;

  // Prologue: stage group 0 tile into buf[0].
  {
    const float* xc = xn + (size_t)wc * H * W;
    async_ld_f32(xc + gofs[0], &buf[0][lofs[0]]);
    async_ld_f32(xc + gofs[1], &buf[0][lofs[1]]);
  }

  for (int g = 0; g < G; ++g) {
    if (g + 1 < G) {
      // Prefetch next group's tile into the other buffer, then wait for the
      // current tile (the 2 newest async ops are the only ones outstanding).
      const float* xc = xn + (size_t)((g + 1) * WC + wc) * H * W;
      float* b = buf[(g + 1) & 1];
      async_ld_f32(xc + gofs[0], &b[lofs[0]]);
      async_ld_f32(xc + gofs[1], &b[lofs[1]]);
      WAIT_ASYNC(2);
    } else {
      WAIT_ASYNC(0);
    }
    __syncthreads();  // all waves' tile-g loads are now in LDS

    const float* b = buf[g & 1];
    float acc = 0.0f;
#pragma unroll
    for (int kh = 0; kh < 3; ++kh)
#pragma unroll
      for (int kw = 0; kw < 3; ++kw)
        acc = fmaf(b[(ty + kh) * LSTRIDE + tx + kw], wt[kh * 3 + kw], acc);

    out[((size_t)(n * C + g * WC + wc) * H + h) * W + w] = acc;
    __syncthreads();  // reads of buf[g&1] done before it is overwritten
  }
}

extern "C" void kernel_launch(void* const* d_in, const int* in_sizes, int n_in,
                              void* d_out, int out_size, void* d_ws,
                              size_t ws_size, hipStream_t stream) {
  using namespace cfg;
  const float* x = (const float*)d_in[0];
  const float* wgt = (const float*)d_in[1];
  float* out = (float*)d_out;

  dim3 grid((H / TH) * (W / TW), WC, N);  // 64 x 32 x 8
  dim3 block(TW, TH, 1);                  // 256 threads = 8 waves
  aggregation_kernel<<<grid, block, 0, stream>>>(x, wgt, out);
}